// HFDE_77292231458936
// MI455X (gfx1250) — compile-verified
//
#include <hip/hip_runtime.h>

// Problem constants (from reference setup_inputs)
#define B_    4
#define C_    256
#define H_    256
#define W_    256
#define HW_   (H_ * W_)
#define COMP_ 64
#define KK_   9

typedef __attribute__((ext_vector_type(2))) float v2f;
typedef __attribute__((ext_vector_type(8))) float v8f;

// ---------------------------------------------------------------------------
// Kernel 1: 1x1 channel compression as GEMM via V_WMMA_F32_16X16X4_F32.
// Per batch: comp(64 x 65536) = w_comp(64 x 256) * x(256 x 65536).
// One wave owns a 16(o) x 64(p) super-tile: 4 independent accumulator chains
// (pipelines the XDL unit) and the A fragment is loaded once per K-step for
// 4 WMMAs. K=256 swept in steps of 4. Layouts per CDNA5 ISA 7.12.2.
// ---------------------------------------------------------------------------
__global__ __launch_bounds__(256) void k1_compress(const float* __restrict__ x,
                                                   const float* __restrict__ wc,
                                                   float* __restrict__ comp)
{
    const int wave = blockIdx.x * 8 + (threadIdx.x >> 5);
    const int lane = threadIdx.x & 31;
    const int l16  = lane & 15;
    const int hi   = lane >> 4;               // 0 = lanes 0-15, 1 = lanes 16-31

    const int tilesN = HW_ / 64;              // 1024 N-supertiles (64 wide)
    const int tilesPerBatch = (COMP_ / 16) * tilesN;   // 4096
    const int b  = wave / tilesPerBatch;
    const int t  = wave % tilesPerBatch;
    const int mt = t / tilesN;                // o-tile
    const int nt = t % tilesN;                // p-supertile
    const int o0 = mt * 16;
    const int p0 = nt * 64;

    const float* xb = x  + (size_t)b * C_ * HW_ + p0 + l16;  // column base
    const float* wr = wc + (size_t)(o0 + l16) * C_;          // row o0+l16 of w_comp

    v8f acc0 = {}, acc1 = {}, acc2 = {}, acc3 = {};
#pragma unroll 4
    for (int k = 0; k < C_; k += 4) {
        const int ka = k + 2 * hi;            // A/B VGPR0 holds K={0,2}, VGPR1 K={1,3}
        v2f a;                                // A: M=l16, contiguous c pair -> b64 load
        a.x = wr[ka];
        a.y = wr[ka + 1];
        const float* xr0 = xb + (size_t)ka * HW_;
        const float* xr1 = xb + (size_t)(ka + 1) * HW_;
        v2f b0; b0.x = xr0[0];  b0.y = xr1[0];
        v2f b1; b1.x = xr0[16]; b1.y = xr1[16];
        v2f b2; b2.x = xr0[32]; b2.y = xr1[32];
        v2f b3; b3.x = xr0[48]; b3.y = xr1[48];
        acc0 = __builtin_amdgcn_wmma_f32_16x16x4_f32(false, a, false, b0, (short)0, acc0, false, false);
        acc1 = __builtin_amdgcn_wmma_f32_16x16x4_f32(false, a, false, b1, (short)0, acc1, false, false);
        acc2 = __builtin_amdgcn_wmma_f32_16x16x4_f32(false, a, false, b2, (short)0, acc2, false, false);
        acc3 = __builtin_amdgcn_wmma_f32_16x16x4_f32(false, a, false, b3, (short)0, acc3, false, false);
    }

    // C/D layout: VGPR r -> M = r + 8*hi, N = l16 (+16/32/48 per sub-tile)
    float* cb = comp + (size_t)b * COMP_ * HW_ + (size_t)(o0 + 8 * hi) * HW_ + p0 + l16;
#pragma unroll
    for (int r = 0; r < 8; ++r) {
        cb[(size_t)r * HW_ + 0]  = acc0[r];
        cb[(size_t)r * HW_ + 16] = acc1[r];
        cb[(size_t)r * HW_ + 32] = acc2[r];
        cb[(size_t)r * HW_ + 48] = acc3[r];
    }
}

// ---------------------------------------------------------------------------
// Kernel 2: 3x3 conv (64 -> 9, SAME zero pad) + bias + softmax over the 9 taps.
// Block = one (b, h) row, thread = w. LDS stages 16 channels x 3 rows x (W+2).
// Weights are thread-uniform -> scalar (s_load) path through constant cache.
// ---------------------------------------------------------------------------
#define CCH 16
__global__ __launch_bounds__(256) void k2_conv_softmax(const float* __restrict__ comp,
                                                       const float* __restrict__ wg,
                                                       const float* __restrict__ bg,
                                                       float* __restrict__ sm)
{
    __shared__ float lds[CCH * 3 * (W_ + 2)];   // 49,536 B

    const int b = blockIdx.x / H_;
    const int h = blockIdx.x % H_;
    const int w = threadIdx.x;

    float lg[KK_];
#pragma unroll
    for (int kk = 0; kk < KK_; ++kk) lg[kk] = bg[kk];

    for (int cc = 0; cc < COMP_; cc += CCH) {
        // Cooperative, zero-padded stage of CCH channels x 3 rows x (W+2) cols.
        for (int i = threadIdx.x; i < CCH * 3 * (W_ + 2); i += 256) {
            const int ch  = i / (3 * (W_ + 2));
            const int rem = i % (3 * (W_ + 2));
            const int r   = rem / (W_ + 2);
            const int col = rem % (W_ + 2);
            const int hh  = h + r - 1;
            const int ww  = col - 1;
            float v = 0.0f;
            if (hh >= 0 && hh < H_ && ww >= 0 && ww < W_)
                v = comp[((size_t)(b * COMP_ + cc + ch) * H_ + hh) * W_ + ww];
            lds[i] = v;
        }
        __syncthreads();

        for (int ch = 0; ch < CCH; ++ch) {
            float v[9];
#pragma unroll
            for (int r = 0; r < 3; ++r)
#pragma unroll
                for (int dj = 0; dj < 3; ++dj)
                    v[r * 3 + dj] = lds[(ch * 3 + r) * (W_ + 2) + w + dj];
#pragma unroll
            for (int kk = 0; kk < KK_; ++kk) {
                const float* wk = wg + ((size_t)kk * COMP_ + (cc + ch)) * 9;
#pragma unroll
                for (int t = 0; t < 9; ++t)
                    lg[kk] = fmaf(wk[t], v[t], lg[kk]);
            }
        }
        __syncthreads();
    }

    // Exact softmax over the 9 logits.
    float m = lg[0];
#pragma unroll
    for (int kk = 1; kk < KK_; ++kk) m = fmaxf(m, lg[kk]);
    float e[KK_], s = 0.0f;
#pragma unroll
    for (int kk = 0; kk < KK_; ++kk) { e[kk] = __expf(lg[kk] - m); s += e[kk]; }
    const float inv = 1.0f / s;

    float* smb = sm + (size_t)b * KK_ * HW_ + (size_t)h * W_ + w;
#pragma unroll
    for (int kk = 0; kk < KK_; ++kk)
        smb[(size_t)kk * HW_] = e[kk] * inv;
}

// ---------------------------------------------------------------------------
// Kernel 3: view-scramble + hamming + renorm + reflect-padded 9-tap aggregate,
// out = 2x - agg. The .view() reinterpret means pixel (h,w) uses the 9
// CONSECUTIVE flat elements of the softmax buffer at offset (h*W+w)*9.
// Vectorized 4-wide in w: per channel-row one b128 + two edge b32 loads and a
// b128 store (9 loads / 4 outputs). kwn (4 pos x 9 taps) held in registers
// across all 64 channel iterations of the thread.
// ---------------------------------------------------------------------------
__global__ __launch_bounds__(256) void k3_aggregate(const float* __restrict__ x,
                                                    const float* __restrict__ sm,
                                                    float* __restrict__ out)
{
    const int b  = blockIdx.x / H_;
    const int h  = blockIdx.x % H_;
    const int wq = threadIdx.x & 63;          // 64 quads cover W = 256
    const int cg = threadIdx.x >> 6;          // 4 channel groups (stride-4 loop)
    const int w0 = wq * 4;

    // hamming(3) = [0.08, 1.0, 0.08]; ham[i*3+j] = hr[i]*hr[j]
    const float ham[9] = { 0.0064f, 0.08f, 0.0064f,
                           0.08f,   1.0f,  0.08f,
                           0.0064f, 0.08f, 0.0064f };

    // Normalized kernel weights for the 4 pixels of this quad.
    const float* s9 = sm + (size_t)b * KK_ * HW_ + (size_t)(h * W_ + w0) * 9;
    float kwn[4][9];
#pragma unroll
    for (int j = 0; j < 4; ++j) {
        float ssum = 1e-8f;
#pragma unroll
        for (int t = 0; t < 9; ++t) { kwn[j][t] = s9[j * 9 + t] * ham[t]; ssum += kwn[j][t]; }
        const float inv = 1.0f / ssum;
#pragma unroll
        for (int t = 0; t < 9; ++t) kwn[j][t] *= inv;
    }

    // Reflect padding (numpy 'reflect': -1 -> 1, N -> N-2)
    const int hm  = (h == 0)      ? 1      : h - 1;
    const int hp  = (h == H_ - 1) ? H_ - 2 : h + 1;
    const int wlo = (w0 == 0)         ? 1      : w0 - 1;   // left halo scalar
    const int whi = (w0 + 4 > W_ - 1) ? W_ - 2 : w0 + 4;   // right halo scalar

    const float* xb = x   + (size_t)b * C_ * HW_;
    float*       ob = out + (size_t)b * C_ * HW_;
    const int rows[3] = { hm, h, hp };

    for (int c = cg; c < C_; c += 4) {
        const float* xc = xb + (size_t)c * HW_;
        if (c + 4 < C_)
            __builtin_prefetch(xc + 4 * HW_ + (size_t)h * W_ + w0, 0, 1);  // global_prefetch_b8

        // win[r][0..5] = x row at columns [w0-1, w0..w0+3, w0+4] (reflected halos)
        float win[3][6];
#pragma unroll
        for (int r = 0; r < 3; ++r) {
            const float* rp = xc + (size_t)rows[r] * W_;
            const float4 mid = *(const float4*)(rp + w0);   // 16B-aligned (w0 % 4 == 0)
            win[r][0] = rp[wlo];
            win[r][1] = mid.x; win[r][2] = mid.y; win[r][3] = mid.z; win[r][4] = mid.w;
            win[r][5] = rp[whi];
        }

        float o[4];
#pragma unroll
        for (int j = 0; j < 4; ++j) {
            float agg = 0.0f;
#pragma unroll
            for (int r = 0; r < 3; ++r)
#pragma unroll
                for (int dj = 0; dj < 3; ++dj)
                    agg = fmaf(kwn[j][r * 3 + dj], win[r][j + dj], agg);
            o[j] = 2.0f * win[1][j + 1] - agg;   // out = 2x - agg (residual highpass)
        }
        float4 res; res.x = o[0]; res.y = o[1]; res.z = o[2]; res.w = o[3];
        *(float4*)(ob + (size_t)c * HW_ + (size_t)h * W_ + w0) = res;
    }
}

// ---------------------------------------------------------------------------
extern "C" void kernel_launch(void* const* d_in, const int* in_sizes, int n_in,
                              void* d_out, int out_size, void* d_ws, size_t ws_size,
                              hipStream_t stream)
{
    (void)in_sizes; (void)n_in; (void)out_size; (void)ws_size;

    const float* x  = (const float*)d_in[0];   // (4,256,256,256) f32
    const float* wc = (const float*)d_in[1];   // (64,256) f32
    const float* wg = (const float*)d_in[2];   // (9,64,3,3) f32
    const float* bg = (const float*)d_in[3];   // (9,) f32
    float* out = (float*)d_out;

    // Workspace: compressed (B*64*HW f32 = 64 MiB) then softmax (B*9*HW = 9 MiB)
    float* comp = (float*)d_ws;
    float* sm   = comp + (size_t)B_ * COMP_ * HW_;

    // K1: B * (64/16) * (HW/64) = 16384 waves -> 2048 blocks x 8 waves
    const int k1_blocks = (B_ * (COMP_ / 16) * (HW_ / 64)) / 8;
    k1_compress<<<dim3(k1_blocks), dim3(256), 0, stream>>>(x, wc, comp);
    k2_conv_softmax<<<dim3(B_ * H_), dim3(256), 0, stream>>>(comp, wg, bg, sm);
    k3_aggregate<<<dim3(B_ * H_), dim3(256), 0, stream>>>(x, sm, out);
}